// smoothTransformer3D_38354057954071
// MI455X (gfx1250) — compile-verified
//
#include <hip/hip_runtime.h>

// CDNA5 / gfx1250 smooth 3D spatial transformer.
//   d_in[0] = im      [2,128,128,128,1] f32
//   d_in[1] = defgrad [2,128,128,128,3] f32
//   d_in[2] = affine  [2,12]            f32
//   d_out   = output (4,194,304 f32) ++ grid (12,582,912 f32)

typedef __attribute__((ext_vector_type(2))) float v2f;
typedef __attribute__((ext_vector_type(8))) float v8f;

#define N128 128

__device__ __forceinline__ float squash(float v) {
    // c / (1 + (c-1)*exp(-v)) with c = 2.0
    return 2.0f / (1.0f + expf(-v));
}

// Axis-wise running-sum of squash(defgrad[..,axis]) into cum[..,axis].
// axis 0: blocks=(b,y), threads=z, loop x   (lanes stride 12B -> 3 lines/wave)
// axis 1: blocks=(b,x), threads=z, loop y
// axis 2: blocks=(b,x), threads=y, loop z   (32 sequential streams per wave)
__global__ void st3d_scan_kernel(const float* __restrict__ dg,
                                 float* __restrict__ cum, int axis) {
    const int blk = blockIdx.x;        // 256 blocks
    const int t   = threadIdx.x;       // 128 threads
    const int b   = blk >> 7;
    const int q   = blk & 127;

    int base, stride;
    if (axis == 0) {        // (b, x=i, y=q, z=t), ch 0
        base   = (((b * N128 + 0) * N128 + q) * N128 + t) * 3 + 0;
        stride = N128 * N128 * 3;
    } else if (axis == 1) { // (b, x=q, y=i, z=t), ch 1
        base   = (((b * N128 + q) * N128 + 0) * N128 + t) * 3 + 1;
        stride = N128 * 3;
    } else {                // (b, x=q, y=t, z=i), ch 2
        base   = (((b * N128 + q) * N128 + t) * N128 + 0) * 3 + 2;
        stride = 3;
    }

    float s = 0.0f;
#pragma unroll 4
    for (int i = 0; i < N128; ++i) {
        s += squash(dg[base]);
        cum[base] = s;
        base += stride;
    }
}

// Fused affine (via V_WMMA_F32_16X16X4_F32) + trilinear sampling.
// Blocks = (b,x,y) lines (32768), 128 threads = z. Each wave owns z in
// [32w, 32w+32): two WMMA tiles of 16 voxels (M=voxel, K={gx,gy,gz,1}),
// B = affine^T (cols>=3 zero), D columns 0..2 routed through LDS.
__global__ void st3d_affine_sample_kernel(const float* __restrict__ im,
                                          const float* __restrict__ affine,
                                          float* __restrict__ out,
                                          float* grid) {
    __shared__ __align__(16) float stage[4 * 96]; // 4 waves x 32 voxels x 3

    const int t    = threadIdx.x;
    const int wave = t >> 5;
    const int lane = t & 31;
    const int half = lane >> 4;   // 0: lanes 0-15, 1: lanes 16-31
    const int m    = lane & 15;   // A row / B column within tile

    const int blk = blockIdx.x;               // (b*128 + x)*128 + y
    const int b   = blk >> 14;
    const int xy  = blk & 16383;
    const int x   = xy >> 7;
    const int y   = xy & 127;
    const int lineBase = ((b * N128 + x) * N128 + y) * N128; // voxel idx, z=0
    const int zb = wave * 32;

    // ---- B operand: B[k][n] = Aaff[n][k], Aaff = affine[b] + I(3x4) ----
    v2f bmat; bmat[0] = 0.0f; bmat[1] = 0.0f;
    if (m < 3) {
        const int k0 = half ? 2 : 0;
        float b0 = affine[b * 12 + m * 4 + k0];
        float b1 = affine[b * 12 + m * 4 + k0 + 1];
        if (m == k0)     b0 += 1.0f;
        if (m == k0 + 1) b1 += 1.0f;
        bmat[0] = b0; bmat[1] = b1;
    }

    // ---- A operands: pre-affine cumsums (staged earlier into `grid`) ----
    const float* cumBase = grid + (lineBase + zb) * 3;
    const float gx0 = cumBase[m * 3 + 0];
    const float gy0 = cumBase[m * 3 + 1];
    const float gz0 = cumBase[m * 3 + 2];
    const float gx1 = cumBase[(16 + m) * 3 + 0];
    const float gy1 = cumBase[(16 + m) * 3 + 1];
    const float gz1 = cumBase[(16 + m) * 3 + 2];

    v2f a0, a1;
    a0[0] = half ? gz0 : gx0;  a0[1] = half ? 1.0f : gy0;
    a1[0] = half ? gz1 : gx1;  a1[1] = half ? 1.0f : gy1;

    v8f cz = {};
    // D(16x16,f32) = A(16x4,f32) x B(4x16,f32) + 0
    v8f d0 = __builtin_amdgcn_wmma_f32_16x16x4_f32(
        false, a0, false, bmat, (short)0, cz, false, false);
    v8f d1 = __builtin_amdgcn_wmma_f32_16x16x4_f32(
        false, a1, false, bmat, (short)0, cz, false, false);

    // ---- scatter D cols 0..2 into voxel-ordered triples in LDS ----
    float* st = &stage[wave * 96];
    if (m < 3) {
#pragma unroll
        for (int r = 0; r < 8; ++r) {
            const int row = r + 8 * half;        // voxel within tile
            st[row * 3 + m]      = d0[r];
            st[48 + row * 3 + m] = d1[r];
        }
    }
    __syncthreads();

    // ---- grid output: 96 contiguous floats per wave, float4 stores ----
    float* gdst = grid + (lineBase + zb) * 3;
    if (lane < 24) {
        const float4 v = *(const float4*)&st[lane * 4];
        *(float4*)&gdst[lane * 4] = v;
    }

    // ---- trilinear sampling for this lane's voxel (z = zb + lane) ----
    const float xs = st[lane * 3 + 0];
    const float ys = st[lane * 3 + 1];
    const float zs = st[lane * 3 + 2];

    const int fx = (int)floorf(xs);
    const int fy = (int)floorf(ys);
    const int fz = (int)floorf(zs);
    const int x0 = min(max(fx, 0), N128 - 1);
    const int x1 = min(max(fx + 1, 0), N128 - 1);
    const int y0 = min(max(fy, 0), N128 - 1);
    const int y1 = min(max(fy + 1, 0), N128 - 1);
    const int z0 = min(max(fz, 0), N128 - 1);
    const int z1 = min(max(fz + 1, 0), N128 - 1);

    const float xd = xs - (float)x0;   // reference uses clipped corner
    const float yd = ys - (float)y0;
    const float zd = zs - (float)z0;

    const float* vol = im + b * (N128 * N128 * N128);
#define AT(xi, yi, zi) vol[((xi) * N128 + (yi)) * N128 + (zi)]
    const float Ia = AT(x0, y0, z0), Ib = AT(x0, y0, z1);
    const float Ic = AT(x0, y1, z0), Id = AT(x0, y1, z1);
    const float Ie = AT(x1, y0, z0), If = AT(x1, y0, z1);
    const float Ig = AT(x1, y1, z0), Ih = AT(x1, y1, z1);
#undef AT

    const float cae   = Ia + (Ie - Ia) * xd;
    const float cbf   = Ib + (If - Ib) * xd;
    const float ccg   = Ic + (Ig - Ic) * xd;
    const float cdh   = Id + (Ih - Id) * xd;
    const float caecg = cae + (ccg - cae) * yd;
    const float cbfdh = cbf + (cdh - cbf) * yd;

    out[lineBase + zb + lane] = caecg + (cbfdh - caecg) * zd;
}

extern "C" void kernel_launch(void* const* d_in, const int* in_sizes, int n_in,
                              void* d_out, int out_size, void* d_ws, size_t ws_size,
                              hipStream_t stream) {
    (void)in_sizes; (void)n_in; (void)out_size; (void)d_ws; (void)ws_size;
    const float* im      = (const float*)d_in[0];
    const float* defgrad = (const float*)d_in[1];
    const float* affine  = (const float*)d_in[2];

    float* out = (float*)d_out;
    const int outElems = 2 * N128 * N128 * N128;  // 4,194,304
    float* grid = out + outElems;                 // 12,582,912 floats

    // Stage pre-affine cumsums into the grid output region (no workspace
    // size assumptions), one kernel per axis; same-stream order suffices.
    st3d_scan_kernel<<<256, 128, 0, stream>>>(defgrad, grid, 0);
    st3d_scan_kernel<<<256, 128, 0, stream>>>(defgrad, grid, 1);
    st3d_scan_kernel<<<256, 128, 0, stream>>>(defgrad, grid, 2);

    // Fused affine (WMMA f32 16x16x4) + trilinear sample; overwrites the
    // staged cumsums in place with the final grid.
    st3d_affine_sample_kernel<<<2 * N128 * N128, 128, 0, stream>>>(
        im, affine, out, grid);
}